// SDFSceneLoss_75376676045581
// MI455X (gfx1250) — compile-verified
//
#include <hip/hip_runtime.h>
#include <math.h>

// ---------------------------------------------------------------------------
// SDF scene loss for MI455X (gfx1250, wave32).
// Compute-bound fp32 workload: all p-linear pairwise terms come from
// V_WMMA_F32_16X16X4_F32 with homogeneous coords [x,y,z,1] (affine constant
// folded into K=4). Per-face denominators are precomputed as reciprocals so
// the 65M-pair inner loop has zero divisions and zero branches; the ray-cast
// test is reduced to sign tests on existing WMMA outputs via
// (xint - p.x) * det == -pn. Face tables + phi (~0.5 MB) live in L2/WGP$.
// ---------------------------------------------------------------------------

typedef float v2f __attribute__((ext_vector_type(2)));
typedef float v4f __attribute__((ext_vector_type(4)));
typedef float v8f __attribute__((ext_vector_type(8)));

#define GRD   32
#define NPTS  32768          // 32^3
#define NF    1000
#define FPAD  1024
#define SFACE (8 * 4 * FPAD) // 8 float4 tables per mesh = 32768 floats
#define OFF_NORM 0           // 2 meshes x (cx,cy,cz,scale)
#define OFF_FACE 16
#define OFF_PHI  (OFF_FACE + 2 * SFACE) // 65552
// total ws floats: OFF_PHI + 2*NPTS = 131088  (~524 KB of d_ws)

__device__ __forceinline__ float gco(int i) {
    return (2.0f * (float)i + 1.0f) * (1.0f / 32.0f) - 1.0f;
}
__device__ __forceinline__ float safe_pos(float x) { return x <= 0.0f ? 1.0f : x; }

// ---------------- kernel 1: bbox -> center/scale per mesh ------------------
__global__ void __launch_bounds__(256)
norm_kernel(const float* __restrict__ v0, const float* __restrict__ v1,
            float* __restrict__ ws) {
    __shared__ float sMn[3 * 256];
    __shared__ float sMx[3 * 256];
    const int tid = threadIdx.x;
    const float* V = blockIdx.x ? v1 : v0;
    float mn[3] = {3.0e38f, 3.0e38f, 3.0e38f};
    float mx[3] = {-3.0e38f, -3.0e38f, -3.0e38f};
    for (int i = tid; i < NF; i += 256) {
        #pragma unroll
        for (int c = 0; c < 3; ++c) {
            float v = V[3 * i + c];
            mn[c] = fminf(mn[c], v);
            mx[c] = fmaxf(mx[c], v);
        }
    }
    #pragma unroll
    for (int c = 0; c < 3; ++c) { sMn[c * 256 + tid] = mn[c]; sMx[c * 256 + tid] = mx[c]; }
    __syncthreads();
    for (int s = 128; s > 0; s >>= 1) {
        if (tid < s) {
            #pragma unroll
            for (int c = 0; c < 3; ++c) {
                sMn[c * 256 + tid] = fminf(sMn[c * 256 + tid], sMn[c * 256 + tid + s]);
                sMx[c * 256 + tid] = fmaxf(sMx[c * 256 + tid], sMx[c * 256 + tid + s]);
            }
        }
        __syncthreads();
    }
    if (tid == 0) {
        float e0 = (sMx[0] - sMn[0]) * 0.6f;   // (1+SCALE_FACTOR)*0.5
        float e1 = (sMx[256] - sMn[256]) * 0.6f;
        float e2 = (sMx[512] - sMn[512]) * 0.6f;
        ws[OFF_NORM + 4 * blockIdx.x + 0] = (sMn[0] + sMx[0]) * 0.5f;
        ws[OFF_NORM + 4 * blockIdx.x + 1] = (sMn[256] + sMx[256]) * 0.5f;
        ws[OFF_NORM + 4 * blockIdx.x + 2] = (sMn[512] + sMx[512]) * 0.5f;
        ws[OFF_NORM + 4 * blockIdx.x + 3] = fmaxf(e0, fmaxf(e1, e2));
    }
}

// ---------------- kernel 2: per-face B-columns + scalars -------------------
// Tables (float4 x FPAD each):
//  0 Bab  = (ab, -ab.a)              4 Bu = (0, ac.z, -ac.y, a.z*ac.y-a.y*ac.z)
//  1 Bac  = (ac, -ac.a)              5 Bv = (0, -ab.z, ab.y, a.y*ab.z-a.z*ab.y)
//  2 Bap  = (-2a, a.a)               6 FS0 = (absq, acsq, abac, 1/safe(nsq))
//  3 Bn   = (n, -n.a)                7 FS1 = (1/safe(absq), 1/safe(acsq),
//                                            1/safe(bcsq), nx)
__global__ void __launch_bounds__(256)
face_kernel(const float* __restrict__ v0, const float* __restrict__ v1,
            const int* __restrict__ f0, const int* __restrict__ f1,
            float* __restrict__ ws) {
    int idx = blockIdx.x * blockDim.x + threadIdx.x;
    if (idx >= 2 * FPAD) return;
    const int mesh = idx >> 10;
    const int f = idx & (FPAD - 1);
    float* FB = ws + OFF_FACE + mesh * SFACE;
    if (f >= NF) {
        v4f z = {};
        #pragma unroll
        for (int k = 0; k < 8; ++k) *(v4f*)(FB + k * 4096 + 4 * f) = z;
        return;
    }
    const float* V = mesh ? v1 : v0;
    const int* FI = mesh ? f1 : f0;
    const float cx = ws[OFF_NORM + 4 * mesh + 0];
    const float cy = ws[OFF_NORM + 4 * mesh + 1];
    const float cz = ws[OFF_NORM + 4 * mesh + 2];
    const float sc = ws[OFF_NORM + 4 * mesh + 3];
    const int ia = FI[3 * f + 0], ib = FI[3 * f + 1], ic = FI[3 * f + 2];
    const float ax = (V[3 * ia + 0] - cx) / sc, ay = (V[3 * ia + 1] - cy) / sc, az = (V[3 * ia + 2] - cz) / sc;
    const float bx = (V[3 * ib + 0] - cx) / sc, by = (V[3 * ib + 1] - cy) / sc, bz = (V[3 * ib + 2] - cz) / sc;
    const float ox = (V[3 * ic + 0] - cx) / sc, oy = (V[3 * ic + 1] - cy) / sc, oz = (V[3 * ic + 2] - cz) / sc;
    const float abx = bx - ax, aby = by - ay, abz = bz - az;
    const float acx = ox - ax, acy = oy - ay, acz = oz - az;
    const float nx = aby * acz - abz * acy;
    const float ny = abz * acx - abx * acz;
    const float nz = abx * acy - aby * acx;
    const float absq = abx * abx + aby * aby + abz * abz;
    const float acsq = acx * acx + acy * acy + acz * acz;
    const float abac = abx * acx + aby * acy + abz * acz;
    const float nsq  = nx * nx + ny * ny + nz * nz;
    const float bcsq = absq + acsq - 2.0f * abac;

    v4f Bab = {abx, aby, abz, -(abx * ax + aby * ay + abz * az)};
    v4f Bac = {acx, acy, acz, -(acx * ax + acy * ay + acz * az)};
    v4f Bap = {-2.0f * ax, -2.0f * ay, -2.0f * az, ax * ax + ay * ay + az * az};
    v4f Bn  = {nx, ny, nz, -(nx * ax + ny * ay + nz * az)};
    v4f Bu  = {0.0f, acz, -acy, az * acy - ay * acz};   // u*det affine in p
    v4f Bv  = {0.0f, -abz, aby, ay * abz - az * aby};   // v*det affine in p
    v4f FS0 = {absq, acsq, abac, 1.0f / safe_pos(nsq)};
    v4f FS1 = {1.0f / safe_pos(absq), 1.0f / safe_pos(acsq),
               1.0f / safe_pos(bcsq), nx};
    *(v4f*)(FB + 0 * 4096 + 4 * f) = Bab;
    *(v4f*)(FB + 1 * 4096 + 4 * f) = Bac;
    *(v4f*)(FB + 2 * 4096 + 4 * f) = Bap;
    *(v4f*)(FB + 3 * 4096 + 4 * f) = Bn;
    *(v4f*)(FB + 4 * 4096 + 4 * f) = Bu;
    *(v4f*)(FB + 5 * 4096 + 4 * f) = Bv;
    *(v4f*)(FB + 6 * 4096 + 4 * f) = FS0;
    *(v4f*)(FB + 7 * 4096 + 4 * f) = FS1;
}

// ---------------- kernel 3: WMMA tile kernel (phi) -------------------------
// One wave32 per 16-point tile; 6x V_WMMA_F32_16X16X4_F32 per 16-face chunk.
// D layout: lane (t,h) slot r holds (M = r+8h, N = t). Inner loop is
// division-free and branch-free.
__device__ __forceinline__ v8f wmma4(v2f a, v2f b, v8f c) {
    return __builtin_amdgcn_wmma_f32_16x16x4_f32(false, a, false, b, (short)0, c,
                                                 false, false);
}

__global__ void __launch_bounds__(256)
phi_kernel(float* __restrict__ ws) {
    const int lane = threadIdx.x & 31;
    const int wave = blockIdx.x * (blockDim.x >> 5) + (threadIdx.x >> 5);
    const int mesh = wave >> 11;      // 2048 tiles per mesh
    const int tile = wave & 2047;
    const int tbase = tile << 4;
    const int t = lane & 15;
    const int h = lane >> 4;

    // A operand: row M = tbase+t; K = (x,y) on low half, (z,1) on high half
    v2f A;
    {
        const int pm = tbase + t;
        const float pmx = gco(pm & 31), pmy = gco((pm >> 5) & 31), pmz = gco(pm >> 10);
        A.x = h ? pmz : pmx;
        A.y = h ? 1.0f : pmy;
    }
    // per-slot point data (points M = r + 8h of this tile)
    float psq8[8], px8[8];
    #pragma unroll
    for (int r = 0; r < 8; ++r) {
        const int pi = tbase + r + 8 * h;
        const float x = gco(pi & 31), y = gco((pi >> 5) & 31), z = gco(pi >> 10);
        psq8[r] = x * x + y * y + z * z;
        px8[r] = x;   // kept for clarity; ray test now uses pn sign only
    }
    float minq[8];
    int cnt[8];
    #pragma unroll
    for (int r = 0; r < 8; ++r) { minq[r] = 3.0e38f; cnt[r] = 0; }

    const float* FB = ws + OFF_FACE + mesh * SFACE;
    for (int cb = 0; cb < 1008; cb += 16) {   // 63 chunks cover 1000 faces
        const int fc = cb + t;
        // prefetch next chunk's face-table lines (global_prefetch_b8)
        if (cb + 16 < 1008) {
            __builtin_prefetch(FB + 0 * 4096 + 4 * (fc + 16), 0, 3);
            __builtin_prefetch(FB + 3 * 4096 + 4 * (fc + 16), 0, 3);
            __builtin_prefetch(FB + 6 * 4096 + 4 * (fc + 16), 0, 3);
        }
        const v2f Bab = *(const v2f*)(FB + 0 * 4096 + 4 * fc + 2 * h);
        const v2f Bac = *(const v2f*)(FB + 1 * 4096 + 4 * fc + 2 * h);
        const v2f Bap = *(const v2f*)(FB + 2 * 4096 + 4 * fc + 2 * h);
        const v2f Bn  = *(const v2f*)(FB + 3 * 4096 + 4 * fc + 2 * h);
        const v2f Bu  = *(const v2f*)(FB + 4 * 4096 + 4 * fc + 2 * h);
        const v2f Bv  = *(const v2f*)(FB + 5 * 4096 + 4 * fc + 2 * h);
        v8f zc = {};
        const v8f Dab = wmma4(A, Bab, zc);   // d1 = p.ab - ab.a
        const v8f Dac = wmma4(A, Bac, zc);   // d2
        const v8f Dap = wmma4(A, Bap, zc);   // -2p.a + a.a
        const v8f Dn  = wmma4(A, Bn,  zc);   // pn  (== -(xint-p.x)*det)
        const v8f Du  = wmma4(A, Bu,  zc);   // u*det
        const v8f Dv  = wmma4(A, Bv,  zc);   // v*det

        const v4f fs0 = *(const v4f*)(FB + 6 * 4096 + 4 * fc);
        const v4f fs1 = *(const v4f*)(FB + 7 * 4096 + 4 * fc);
        const float absq = fs0.x, acsq = fs0.y, abac = fs0.z, r_n = fs0.w;
        const float r_ab = fs1.x, r_ac = fs1.y, r_bc = fs1.z, nxf = fs1.w;
        const bool fvalid = fc < NF;
        const bool nvalid = fabsf(nxf) >= 1e-8f;
        const float det = nvalid ? nxf : 1.0f;
        const float sd = (det < 0.0f) ? -1.0f : 1.0f;   // sign(det)
        const bool rayface = nvalid && fvalid;

        #pragma unroll
        for (int r = 0; r < 8; ++r) {
            const float d1 = Dab[r], d2 = Dac[r];
            const float apsq = Dap[r] + psq8[r];
            const float pn = Dn[r];
            const float d3 = d1 - absq, d4 = d2 - abac;
            const float d5 = d1 - abac, d6 = d2 - acsq;
            const float bpsq = apsq - 2.0f * d1 + absq;
            const float cpsq = apsq - 2.0f * d2 + acsq;
            const float e = d4 - d3;
            const float va = d3 * d6 - d5 * d4;
            const float vb = d5 * d2 - d1 * d6;
            const float vc = d1 * d4 - d3 * d2;
            float dq = (pn * pn) * r_n;                                   // face
            dq = (va <= 0.0f && e >= 0.0f && d5 >= d6) ? (bpsq - (e * e) * r_bc) : dq;
            dq = (vb <= 0.0f && d2 >= 0.0f && d6 <= 0.0f) ? (apsq - (d2 * d2) * r_ac) : dq;
            dq = (vc <= 0.0f && d1 >= 0.0f && d3 <= 0.0f) ? (apsq - (d1 * d1) * r_ab) : dq;
            dq = (d6 >= 0.0f && d5 <= d6) ? cpsq : dq;
            dq = (d3 >= 0.0f && d4 <= d3) ? bpsq : dq;
            dq = (d1 <= 0.0f && d2 <= 0.0f) ? apsq : dq;
            minq[r] = fminf(minq[r], fvalid ? dq : 3.0e38f);
            // ray cast (+x) parity: pure sign tests, division-free
            const float su = sd * Du[r];
            const float sv = sd * Dv[r];
            const float ss = sd * (Du[r] + Dv[r] - det);
            const float sp = sd * pn;
            const bool hit = rayface && (su >= 0.0f) && (sv >= 0.0f) &&
                             (ss <= 0.0f) && (sp < 0.0f);
            cnt[r] += hit ? 1 : 0;
        }
    }
    // reduce across the 16 lanes of each half (same point rows)
    #pragma unroll
    for (int r = 0; r < 8; ++r) {
        float m = minq[r];
        int c = cnt[r];
        #pragma unroll
        for (int s = 8; s >= 1; s >>= 1) {
            m = fminf(m, __shfl_xor(m, s, 32));
            c += __shfl_xor(c, s, 32);
        }
        minq[r] = m;
        cnt[r] = c;
    }
    if (t == 0) {  // lane 0 writes M=0..7, lane 16 writes M=8..15
        float* phi = ws + OFF_PHI + mesh * NPTS + tbase + 8 * h;
        #pragma unroll
        for (int r = 0; r < 8; ++r) {
            const float dist = sqrtf(fmaxf(minq[r], 0.0f));
            phi[r] = (cnt[r] & 1) ? dist : 0.0f;
        }
    }
}

// ---------------- kernel 4: trilinear cross-sample + loss ------------------
__global__ void __launch_bounds__(256)
loss_kernel(const float* __restrict__ v0, const float* __restrict__ v1,
            const float* __restrict__ ws, float* __restrict__ out) {
    __shared__ float red[256];
    const int tid = threadIdx.x;
    float acc = 0.0f;
    for (int idx = tid; idx < 2 * NF; idx += 256) {
        const int i1 = (idx >= NF) ? 1 : 0;      // phi source mesh
        const int j = idx - i1 * NF;
        const float* V = (1 - i1) ? v1 : v0;     // verts of the other mesh
        const float cx = ws[OFF_NORM + 4 * i1 + 0];
        const float cy = ws[OFF_NORM + 4 * i1 + 1];
        const float cz = ws[OFF_NORM + 4 * i1 + 2];
        const float sc = ws[OFF_NORM + 4 * i1 + 3];
        const float x = (V[3 * j + 0] - cx) / sc;
        const float y = (V[3 * j + 1] - cy) / sc;
        const float z = (V[3 * j + 2] - cz) / sc;
        const float* vol = ws + OFF_PHI + i1 * NPTS;
        const float px = ((x + 1.0f) * 32.0f - 1.0f) * 0.5f;
        const float py = ((y + 1.0f) * 32.0f - 1.0f) * 0.5f;
        const float pz = ((z + 1.0f) * 32.0f - 1.0f) * 0.5f;
        const float fx = floorf(px), fy = floorf(py), fz = floorf(pz);
        const int x0 = (int)fx, y0 = (int)fy, z0 = (int)fz;
        const float wx = px - fx, wy = py - fy, wz = pz - fz;
        #pragma unroll
        for (int dz = 0; dz < 2; ++dz)
            #pragma unroll
            for (int dy = 0; dy < 2; ++dy)
                #pragma unroll
                for (int dx = 0; dx < 2; ++dx) {
                    const int ix = x0 + dx, iy = y0 + dy, iz = z0 + dz;
                    const bool ok = ((unsigned)ix < 32u) && ((unsigned)iy < 32u) &&
                                    ((unsigned)iz < 32u);
                    const int cxi = ix < 0 ? 0 : (ix > 31 ? 31 : ix);
                    const int cyi = iy < 0 ? 0 : (iy > 31 ? 31 : iy);
                    const int czi = iz < 0 ? 0 : (iz > 31 ? 31 : iz);
                    const float val = ok ? vol[(czi << 10) + (cyi << 5) + cxi] : 0.0f;
                    const float w = (dz ? wz : 1.0f - wz) * (dy ? wy : 1.0f - wy) *
                                    (dx ? wx : 1.0f - wx);
                    acc += w * val;
                }
    }
    red[tid] = acc;
    __syncthreads();
    for (int s = 128; s > 0; s >>= 1) {
        if (tid < s) red[tid] += red[tid + s];
        __syncthreads();
    }
    if (tid == 0) out[0] = red[0];
}

// ---------------------------------------------------------------------------
extern "C" void kernel_launch(void* const* d_in, const int* in_sizes, int n_in,
                              void* d_out, int out_size, void* d_ws, size_t ws_size,
                              hipStream_t stream) {
    (void)in_sizes; (void)n_in; (void)out_size; (void)ws_size;
    const float* v0 = (const float*)d_in[0];  // verts0 [1,1000,3]
    const float* v1 = (const float*)d_in[1];  // verts1 [1,1000,3]
    const int* f0 = (const int*)d_in[2];      // faces0 [1000,3]
    const int* f1 = (const int*)d_in[3];      // faces1 [1000,3]
    float* ws = (float*)d_ws;                 // needs >= 524352 bytes
    float* out = (float*)d_out;               // scalar f32 loss

    norm_kernel<<<2, 256, 0, stream>>>(v0, v1, ws);
    face_kernel<<<(2 * FPAD + 255) / 256, 256, 0, stream>>>(v0, v1, f0, f1, ws);
    phi_kernel<<<512, 256, 0, stream>>>(ws);   // 4096 waves = 2 meshes x 2048 tiles
    loss_kernel<<<1, 256, 0, stream>>>(v0, v1, ws, out);
}